// LSTM_17197049053469
// MI455X (gfx1250) — compile-verified
//
#include <hip/hip_runtime.h>
#include <hip/hip_bf16.h>

typedef __attribute__((ext_vector_type(2))) float v2f;
typedef __attribute__((ext_vector_type(8))) float v8f;

#define LSTM_B   32
#define LSTM_T   2048
#define LSTM_I   512
#define LSTM_H   512
#define LSTM_G4H 2048     // 4*H
#define NWG      32       // workgroups in the persistent scan kernel
#define U_STRIDE 516      // padded row stride (floats) for the LDS U slice

__device__ __forceinline__ v8f wmma_f32_k4(v2f a, v2f b, v8f c) {
    // D = A(16x4) * B(4x16) + C, exact fp32
    return __builtin_amdgcn_wmma_f32_16x16x4_f32(false, a, false, b, (short)0, c,
                                                 false, false);
}

__device__ __forceinline__ float sigmoidf_(float x) {
    return 1.0f / (1.0f + expf(-x));
}

// ---------------------------------------------------------------------------
// Zero h double-buffer (buffer 0) and the global barrier counter every launch
// so graph replay is deterministic.
// ---------------------------------------------------------------------------
__global__ void init_ws(float* hbuf0, unsigned* bar) {
    int i = blockIdx.x * 256 + threadIdx.x;
    if (i < LSTM_B * LSTM_H) hbuf0[i] = 0.0f;
    if (i == 0) *bar = 0u;
}

// ---------------------------------------------------------------------------
// Phase 1: xW[r, g] = sum_k x[r, k] * W_w[g, k] + W_b[g]
// r = b*T + t (65536 rows), g in [0, 2048). One 16x16 tile per wave,
// two independent WMMA accumulator chains over K.
// ---------------------------------------------------------------------------
__global__ void xw_gemm(const float* __restrict__ x,
                        const float* __restrict__ Ww,
                        const float* __restrict__ Wb,
                        float* __restrict__ xw) {
    const int lane  = threadIdx.x & 31;
    const int wave  = threadIdx.x >> 5;
    const int tile  = blockIdx.x * 8 + wave;
    const int ntile = tile & 127;          // 2048/16 column tiles
    const int mtile = tile >> 7;
    const int m0 = mtile * 16;
    const int n0 = ntile * 16;

    const int r     = lane & 15;           // M (for A) / N (for B) within tile
    const int khalf = (lane >> 4) << 1;    // K sub-offset 0 or 2 per ISA layout

    const float* arow = x  + (size_t)(m0 + r) * LSTM_I;   // x row (b,t)
    const float* brow = Ww + (size_t)(n0 + r) * LSTM_I;   // W_w row g (B = W^T)

    v8f acc0 = {}, acc1 = {};
#pragma unroll 4
    for (int k = 0; k < LSTM_I; k += 8) {
        v2f a0 = *(const v2f*)(arow + k + khalf);
        v2f b0 = *(const v2f*)(brow + k + khalf);
        acc0 = wmma_f32_k4(a0, b0, acc0);
        v2f a1 = *(const v2f*)(arow + k + 4 + khalf);
        v2f b1 = *(const v2f*)(brow + k + 4 + khalf);
        acc1 = wmma_f32_k4(a1, b1, acc1);
    }

    const int   n     = lane & 15;
    const float bias  = Wb[n0 + n];
    const int   mbase = m0 + ((lane >> 4) << 3);   // C layout: M = v + 8*(lane/16)
#pragma unroll
    for (int v = 0; v < 8; ++v)
        xw[(size_t)(mbase + v) * LSTM_G4H + n0 + n] = acc0[v] + acc1[v] + bias;
}

// ---------------------------------------------------------------------------
// Phase 2: persistent recurrent scan. 32 WGs x 256 threads (8 wave32).
// WG wg owns hidden columns j in [wg*16, wg*16+16).
// Wave w: gate = w&3 (i,f,g,o), mh = w>>2 (batch rows 0-15 / 16-31).
// U_w slice (128 KB) is staged into LDS once via GLOBAL_LOAD_ASYNC_TO_LDS_B128
// and reused for all 2048 steps. Grid-wide barrier between steps.
// ---------------------------------------------------------------------------
__global__ void lstm_scan(const float* __restrict__ x,
                          const float* __restrict__ Ww,
                          const float* __restrict__ Wb,
                          const float* __restrict__ Uw,
                          const float* __restrict__ Vi,
                          const float* __restrict__ Vf,
                          const float* __restrict__ Vo,
                          const float* __restrict__ xw,   // nullptr => fused
                          float* __restrict__ hseq,
                          float* __restrict__ hT,
                          float* __restrict__ cT,
                          float* hbuf0, float* hbuf1,
                          unsigned* bar, int fused) {
    __shared__ float lds_g[4][32][16];   // gate tiles [gate][batch][jn]
    __shared__ float cst[32][16];        // cell state slice
    extern __shared__ float ldsU[];      // [64][U_STRIDE] U_w slice, 129 KB

    const int tid  = threadIdx.x;
    const int lane = tid & 31;
    const int wave = tid >> 5;
    const int gate = wave & 3;
    const int mh   = wave >> 2;
    const int j0   = blockIdx.x * 16;

    const int r      = lane & 15;
    const int khalf  = (lane >> 4) << 1;
    const int brow_b = mh * 16 + r;                       // batch row for A
    const int ncol   = lane & 15;
    const int gc     = gate * LSTM_H + j0 + ncol;         // row of Ww / Uw
    const int mb0    = mh * 16 + ((lane >> 4) << 3);      // C-tile batch base

    const float* wrow = Ww + (size_t)gc * LSTM_I;

    // ---- stage this WG's U_w slice into LDS with async 128-bit copies ----
    for (int c = tid; c < 64 * 128; c += 256) {           // 8192 x 16B chunks
        const int row = c >> 7;                           // gate*16 + jn
        const int kc  = (c & 127) << 2;                   // k offset (floats)
        const int g   = row >> 4;
        const int jn  = row & 15;
        const float* src = Uw + ((size_t)(g * LSTM_H + j0 + jn) * LSTM_H + kc);
        unsigned dst = (unsigned)(unsigned long long)(const void*)&ldsU[row * U_STRIDE + kc];
        asm volatile("global_load_async_to_lds_b128 %0, %1, off"
                     :: "v"(dst), "v"(src) : "memory");
    }
    asm volatile("s_wait_asynccnt 0x0" ::: "memory");

    for (int p = tid; p < 512; p += 256) cst[p >> 4][p & 15] = 0.0f;
    __syncthreads();

    const float* urow = &ldsU[(size_t)(gate * 16 + ncol) * U_STRIDE];

    for (int t = 0; t < LSTM_T; ++t) {
        const float* hprev = (t & 1) ? hbuf1 : hbuf0;
        float*       hnext = (t & 1) ? hbuf0 : hbuf1;

        v8f acc0, acc1 = {};
        if (fused) {
            acc0 = (v8f){};
            const float* arow = x + ((size_t)brow_b * LSTM_T + t) * LSTM_I;
            __builtin_prefetch(arow + LSTM_I, 0, 0);      // next timestep's row
#pragma unroll 4
            for (int k = 0; k < LSTM_I; k += 8) {
                v2f a0 = *(const v2f*)(arow + k + khalf);
                v2f b0 = *(const v2f*)(wrow + k + khalf);
                acc0 = wmma_f32_k4(a0, b0, acc0);
                v2f a1 = *(const v2f*)(arow + k + 4 + khalf);
                v2f b1 = *(const v2f*)(wrow + k + 4 + khalf);
                acc1 = wmma_f32_k4(a1, b1, acc1);
            }
        } else {
            // preload precomputed xW tile (bias included) into accumulator 0
#pragma unroll
            for (int v = 0; v < 8; ++v)
                acc0[v] = xw[((size_t)(mb0 + v) * LSTM_T + t) * LSTM_G4H + gc];
        }

        const float* hrow = hprev + (size_t)brow_b * LSTM_H;
#pragma unroll 4
        for (int k = 0; k < LSTM_H; k += 8) {
            v2f a0 = *(const v2f*)(hrow + k + khalf);
            v2f b0 = *(const v2f*)(urow + k + khalf);     // ds_load from LDS
            acc0 = wmma_f32_k4(a0, b0, acc0);
            v2f a1 = *(const v2f*)(hrow + k + 4 + khalf);
            v2f b1 = *(const v2f*)(urow + k + 4 + khalf);
            acc1 = wmma_f32_k4(a1, b1, acc1);
        }

        const float bias = fused ? Wb[gc] : 0.0f;
#pragma unroll
        for (int v = 0; v < 8; ++v)
            lds_g[gate][mb0 + v][ncol] = acc0[v] + acc1[v] + bias;
        __syncthreads();

        for (int p = tid; p < 512; p += 256) {
            const int b = p >> 4, jn = p & 15;
            const float ig = lds_g[0][b][jn];
            const float fg = lds_g[1][b][jn];
            const float gg = lds_g[2][b][jn];
            const float og = lds_g[3][b][jn];
            const float c_old = cst[b][jn];
            const float i_t = sigmoidf_(ig + Vi[j0 + jn] * c_old);
            const float f_t = sigmoidf_(fg + Vf[j0 + jn] * c_old);
            const float g_t = tanhf(gg);
            const float c_new = f_t * c_old + i_t * g_t;
            const float o_t = sigmoidf_(og + Vo[j0 + jn] * c_new);
            const float h_new = o_t * tanhf(c_new);
            cst[b][jn] = c_new;
            hnext[b * LSTM_H + j0 + jn] = h_new;
            hseq[((size_t)b * LSTM_T + t) * LSTM_H + j0 + jn] = h_new;
        }

        __threadfence();          // release h writes to device scope
        __syncthreads();
        if (tid == 0) {
            atomicAdd(bar, 1u);
            const unsigned target = (unsigned)(t + 1) * NWG;
            while (__hip_atomic_load(bar, __ATOMIC_RELAXED,
                                     __HIP_MEMORY_SCOPE_AGENT) < target)
                __builtin_amdgcn_s_sleep(1);
        }
        __syncthreads();
        __threadfence();          // acquire: drop stale cached h lines
    }

    // final h (in hbuf0 after even T) and final c
    for (int p = tid; p < 512; p += 256) {
        const int b = p >> 4, jn = p & 15;
        hT[b * LSTM_H + j0 + jn] = hbuf0[b * LSTM_H + j0 + jn];
        cT[b * LSTM_H + j0 + jn] = cst[b][jn];
    }
}

// ---------------------------------------------------------------------------
extern "C" void kernel_launch(void* const* d_in, const int* in_sizes, int n_in,
                              void* d_out, int out_size, void* d_ws, size_t ws_size,
                              hipStream_t stream) {
    const float* x  = (const float*)d_in[0];
    const float* Ww = (const float*)d_in[1];
    const float* Wb = (const float*)d_in[2];
    const float* Uw = (const float*)d_in[3];
    const float* Vi = (const float*)d_in[4];
    const float* Vf = (const float*)d_in[5];
    const float* Vo = (const float*)d_in[6];

    float* out  = (float*)d_out;
    float* hseq = out;                                          // [B,T,H]
    float* hT   = out + (size_t)LSTM_B * LSTM_T * LSTM_H;       // [B,H]
    float* cT   = hT + (size_t)LSTM_B * LSTM_H;                 // [B,H]

    char* ws = (char*)d_ws;
    unsigned* bar = (unsigned*)ws;                              // 256 B slot
    float* hbuf0  = (float*)(ws + 256);
    float* hbuf1  = hbuf0 + (size_t)LSTM_B * LSTM_H;
    size_t fixed  = 256 + 2 * (size_t)LSTM_B * LSTM_H * sizeof(float);
    size_t xw_off = (fixed + 255) & ~(size_t)255;
    size_t xw_bytes = (size_t)LSTM_B * LSTM_T * LSTM_G4H * sizeof(float); // 512 MiB

    const int use_xw = (ws_size >= xw_off + xw_bytes) ? 1 : 0;
    float* xw = use_xw ? (float*)(ws + xw_off) : nullptr;

    const size_t dyn_lds = (size_t)64 * U_STRIDE * sizeof(float);   // ~129 KB

    init_ws<<<(LSTM_B * LSTM_H + 255) / 256, 256, 0, stream>>>(hbuf0, bar);
    if (use_xw) {
        // total tiles = (B*T/16)*(4H/16) = 4096*128; 8 waves per block
        xw_gemm<<<(4096 * 128) / 8, 256, 0, stream>>>(x, Ww, Wb, xw);
    }
    lstm_scan<<<NWG, 256, dyn_lds, stream>>>(x, Ww, Wb, Uw, Vi, Vf, Vo, xw,
                                             hseq, hT, cT, hbuf0, hbuf1, bar,
                                             use_xw ? 0 : 1);
}